// TactileSurfaceLoss_24927990186053
// MI455X (gfx1250) — compile-verified
//
#include <hip/hip_runtime.h>
#include <math.h>

typedef __attribute__((ext_vector_type(2))) float v2f;
typedef __attribute__((ext_vector_type(8))) float v8f;

#define EPS 1e-8f
#define HUBER_DELTA 0.01f
#define CAUCHY_S2 (0.05f*0.05f)
#define GRAD_INV_SIGMA (1.0f/0.02f)
#define CONTACT_THR2 (0.005f*0.005f)

// workspace float layout
#define WS_FMMEAN 8
#define PP 16              // prep partials: q*64+b, q in 0..5 (fm,op,ent,ac,inv,ratio)
#define PA 416             // surface partials: q*64+b, q in 0..2 (huber,normal,contact)
#define PB 608             // knn partials: b (grad)
#define WS_POSF 1024       // float2[2*M], WMMA A-fragment layout:
                           //   entry[tile*32 + r]      = (-2x, -2y)   of row r  (lanes 0..15, K=0,1)
                           //   entry[tile*32 + 16 + r] = (-2z, |p|^2) of row r  (lanes 16..31, K=2,3)
#define WS_NL(M) (1024 + 4*(M))  // float[M]: -log(op+eps)

__device__ __forceinline__ float blockReduceSum(float v) {
  __shared__ float red[32];
  const int lane = threadIdx.x & 31;
  const int w = threadIdx.x >> 5;
  #pragma unroll
  for (int o = 16; o > 0; o >>= 1) v += __shfl_down(v, o, 32);
  __syncthreads();                 // protect red[] across repeated calls
  if (lane == 0) red[w] = v;
  __syncthreads();
  float r = 0.f;
  if (w == 0) {
    r = (lane < (int)(blockDim.x >> 5)) ? red[lane] : 0.f;
    #pragma unroll
    for (int o = 16; o > 0; o >>= 1) r += __shfl_down(r, o, 32);
  }
  return r;  // valid in thread 0
}

__global__ void k_init(float* ws) {
  int t = blockIdx.x * blockDim.x + threadIdx.x;
  if (t < 1024) ws[t] = 0.f;
}

__global__ void __launch_bounds__(256)
k_prep(const float* __restrict__ tf, const float* __restrict__ pos,
       const float* __restrict__ sc, const float* __restrict__ op,
       float* __restrict__ ws, int N, int M)
{
  int t = blockIdx.x * blockDim.x + threadIdx.x;
  float2* posF = (float2*)(ws + WS_POSF);
  float* nl = ws + WS_NL(M);
  float fm = 0.f, opS = 0.f, ent = 0.f, acS = 0.f, invS = 0.f, rat = 0.f;
  if (t < M) {
    float x = pos[3*t], y = pos[3*t+1], z = pos[3*t+2];
    const int tile = t >> 4, r = t & 15;
    posF[tile*32 + r]      = make_float2(-2.f*x, -2.f*y);
    posF[tile*32 + 16 + r] = make_float2(-2.f*z, x*x + y*y + z*z);
    float o = op[t];
    nl[t] = -logf(o + EPS);
    opS = o;
    ent = -(o * logf(o + EPS) + (1.f - o) * logf(1.f - o + EPS));
    float a0 = expf(sc[3*t]), a1 = expf(sc[3*t+1]), a2 = expf(sc[3*t+2]);
    acS = a0 + a1 + a2;
    invS = 1.f/(a0+EPS) + 1.f/(a1+EPS) + 1.f/(a2+EPS);
    rat = fmaxf(a0, fmaxf(a1, a2)) / (fminf(a0, fminf(a1, a2)) + EPS);
  }
  if (t < N) fm = fabsf(tf[t]);
  fm  = blockReduceSum(fm);
  opS = blockReduceSum(opS);
  ent = blockReduceSum(ent);
  acS = blockReduceSum(acS);
  invS = blockReduceSum(invS);
  rat = blockReduceSum(rat);
  if (threadIdx.x == 0) {
    ws[PP + 0*64 + blockIdx.x] = fm;
    ws[PP + 1*64 + blockIdx.x] = opS;
    ws[PP + 2*64 + blockIdx.x] = ent;
    ws[PP + 3*64 + blockIdx.x] = acS;
    ws[PP + 4*64 + blockIdx.x] = invS;
    ws[PP + 5*64 + blockIdx.x] = rat;
  }
}

__global__ void k_fmmean(float* ws, int N, int nb) {
  if (threadIdx.x == 0 && blockIdx.x == 0) {
    float s = 0.f;
    for (int b = 0; b < nb; ++b) s += ws[PP + b];
    ws[WS_FMMEAN] = s / (float)N;
  }
}

// One wave = 16 tactile points (WMMA columns); rows = gaussian positions.
// A[m] = (-2p, |p|^2), B[:,n] = (t, 1), C = |t_n|^2  ->  D[m][n] = d^2(m,n)
__global__ void __launch_bounds__(256)
k_surface(const float* __restrict__ tp, const float* __restrict__ tn,
          const float* __restrict__ tf, const float* __restrict__ pos,
          const float* __restrict__ sc, const float* __restrict__ rot,
          float* __restrict__ ws, int N, int M)
{
  const int lane = threadIdx.x & 31;
  const int wave = threadIdx.x >> 5;
  const int ci = lane & 15;
  const bool hi = lane >= 16;
  const int i = blockIdx.x * 128 + wave * 16 + ci;

  const float2* posF = (const float2*)(ws + WS_POSF);
  const float* nl = ws + WS_NL(M);
  const float fm_mean = ws[WS_FMMEAN];

  float tx = tp[3*i], ty = tp[3*i+1], tz = tp[3*i+2];
  float tpn2 = tx*tx + ty*ty + tz*tz;
  // B fragment (K x N = 4x16): vgpr0 = rows {0,2}, vgpr1 = rows {1,3}, lane&15 = column
  v2f b; b.x = hi ? tz : tx; b.y = hi ? 1.0f : ty;
  // C fragment: |t_n|^2 broadcast over the column (constant per lane)
  v8f cini;
  #pragma unroll
  for (int v = 0; v < 8; ++v) cini[v] = tpn2;

  float bestD = 3.4e38f; int bestI = 0;
  float csum = 0.f; int ccnt = 0;
  const int roff = hi ? 8 : 0;
  const int tiles = M >> 4;

  for (int t = 0; t < tiles; ++t) {
    const int base = t << 4;
    float2 fa = posF[t*32 + lane];          // direct A-fragment load, no select
    __builtin_prefetch(posF + (t + 8)*32 + lane, 0, 3);
    v2f a; a.x = fa.x; a.y = fa.y;
    v8f d = __builtin_amdgcn_wmma_f32_16x16x4_f32(false, a, false, b, (short)0, cini, false, false);
    const float4 n0 = *(const float4*)(nl + base + roff);
    const float4 n1 = *(const float4*)(nl + base + roff + 4);
    float nlv[8] = {n0.x, n0.y, n0.z, n0.w, n1.x, n1.y, n1.z, n1.w};
    // raw d^2 may be epsilon-negative; decisions identical to the clamped value
    #pragma unroll
    for (int v = 0; v < 8; ++v) {
      float d2 = d[v];
      int m = base + roff + v;
      if (d2 < bestD) { bestD = d2; bestI = m; }
      bool hit = d2 < CONTACT_THR2;
      csum += hit ? nlv[v] : 0.f;
      ccnt += hit ? 1 : 0;
    }
  }
  // combine the two lane halves (rows 0..7 vs 8..15)
  float oD = __shfl_xor(bestD, 16, 32);
  int   oI = __shfl_xor(bestI, 16, 32);
  float oC = __shfl_xor(csum, 16, 32);
  int   oN = __shfl_xor(ccnt, 16, 32);
  if (oD < bestD || (oD == bestD && oI < bestI)) { bestD = oD; bestI = oI; }
  csum += oC; ccnt += oN;
  bestD = fmaxf(bestD, 0.f);   // clamp once, before sqrt

  float hub = 0.f, nrm = 0.f, con = 0.f;
  if (!hi) {
    // surface (huber on scale-adaptive nearest distance)
    float a0 = expf(sc[3*bestI]), a1 = expf(sc[3*bestI+1]), a2 = expf(sc[3*bestI+2]);
    float adaptive = (a0 + a1 + a2) * (1.f/3.f);
    float nd = sqrtf(bestD) / (adaptive + EPS);
    hub = (nd < HUBER_DELTA) ? 0.5f*nd*nd : HUBER_DELTA*(nd - 0.5f*HUBER_DELTA);
    // normal (quat -> rot col 2, orient toward point, cauchy)
    float qw = rot[4*bestI], qx = rot[4*bestI+1], qy = rot[4*bestI+2], qz = rot[4*bestI+3];
    float qn = fmaxf(sqrtf(qw*qw + qx*qx + qy*qy + qz*qz), 1e-12f);
    qw /= qn; qx /= qn; qy /= qn; qz /= qn;
    float gx = 2.f*(qx*qz + qw*qy);
    float gy = 2.f*(qy*qz - qw*qx);
    float gz = 1.f - 2.f*(qx*qx + qy*qy);
    float px = pos[3*bestI], py = pos[3*bestI+1], pz = pos[3*bestI+2];
    float dp = gx*(tx-px) + gy*(ty-py) + gz*(tz-pz);
    if (dp < 0.f) { gx = -gx; gy = -gy; gz = -gz; }
    float gl = fmaxf(sqrtf(gx*gx + gy*gy + gz*gz), 1e-12f);
    gx /= gl; gy /= gl; gz /= gl;
    float nx = tn[3*i], ny = tn[3*i+1], nz = tn[3*i+2];
    float err = 1.f - fabsf(nx*gx + ny*gy + nz*gz);
    nrm = -logf(CAUCHY_S2 / (CAUCHY_S2 + err*err) + EPS);
    // contact
    float fm = fabsf(tf[i]);
    float fw = 1.f / (1.f + expf(-fm / fm_mean));
    float rm = (ccnt > 0) ? (csum / fmaxf((float)ccnt, 1.f)) : 0.f;
    con = fw * rm;
  }
  hub = blockReduceSum(hub);
  nrm = blockReduceSum(nrm);
  con = blockReduceSum(con);
  if (threadIdx.x == 0) {
    ws[PA + 0*64 + blockIdx.x] = hub;
    ws[PA + 1*64 + blockIdx.x] = nrm;
    ws[PA + 2*64 + blockIdx.x] = con;
  }
}

// KNN (6 nearest excluding self) + gram/angle loss; same WMMA tiling.
__global__ void __launch_bounds__(256)
k_knn(const float* __restrict__ pos, float* __restrict__ ws, int M)
{
  __shared__ float sMv[8][16][12];
  __shared__ int   sMi[8][16][12];
  const int lane = threadIdx.x & 31;
  const int wave = threadIdx.x >> 5;
  const int ci = lane & 15;
  const bool hi = lane >= 16;
  const int i = blockIdx.x * 128 + wave * 16 + ci;

  const float2* posF = (const float2*)(ws + WS_POSF);
  float cx = pos[3*i], cy = pos[3*i+1], cz = pos[3*i+2];
  float pn2 = cx*cx + cy*cy + cz*cz;
  v2f b; b.x = hi ? cz : cx; b.y = hi ? 1.0f : cy;
  v8f cini;
  #pragma unroll
  for (int v = 0; v < 8; ++v) cini[v] = pn2;

  float lv[6]; int li[6];
  #pragma unroll
  for (int t = 0; t < 6; ++t) { lv[t] = 3.4e38f; li[t] = 0x7fffffff; }

  const int roff = hi ? 8 : 0;
  const int tiles = M >> 4;
  for (int t = 0; t < tiles; ++t) {
    const int base = t << 4;
    float2 fa = posF[t*32 + lane];
    __builtin_prefetch(posF + (t + 8)*32 + lane, 0, 3);
    v2f a; a.x = fa.x; a.y = fa.y;
    v8f d = __builtin_amdgcn_wmma_f32_16x16x4_f32(false, a, false, b, (short)0, cini, false, false);
    #pragma unroll
    for (int v = 0; v < 8; ++v) {
      int m = base + roff + v;
      float d2 = d[v];                       // raw; ordering unaffected by clamp
      if (m != i && d2 < lv[5]) {
        lv[5] = d2; li[5] = m;
        #pragma unroll
        for (int t2 = 5; t2 >= 1; --t2) {
          bool sw = (lv[t2] < lv[t2-1]) || (lv[t2] == lv[t2-1] && li[t2] < li[t2-1]);
          if (sw) {
            float tv = lv[t2]; lv[t2] = lv[t2-1]; lv[t2-1] = tv;
            int ti2 = li[t2]; li[t2] = li[t2-1]; li[t2-1] = ti2;
          }
        }
      }
    }
  }
  // stash both halves' sorted-6 lists, merge in lanes 0..15
  const int lofs = hi ? 6 : 0;
  #pragma unroll
  for (int t = 0; t < 6; ++t) { sMv[wave][ci][lofs + t] = lv[t]; sMi[wave][ci][lofs + t] = li[t]; }
  __syncthreads();

  float grad = 0.f;
  if (!hi) {
    int pa_ = 0, pb_ = 6;
    int nbr[6];
    #pragma unroll
    for (int t = 0; t < 6; ++t) {
      float va = sMv[wave][ci][pa_], vb = sMv[wave][ci][pb_];
      int ia = sMi[wave][ci][pa_], ib = sMi[wave][ci][pb_];
      bool ta = (va < vb) || (va == vb && ia <= ib);
      nbr[t] = ta ? ia : ib;
      if (ta) pa_++; else pb_++;
    }
    float nvx[6], nvy[6], nvz[6];
    #pragma unroll
    for (int t = 0; t < 6; ++t) {
      int j = nbr[t];
      float vx = pos[3*j] - cx, vy = pos[3*j+1] - cy, vz = pos[3*j+2] - cz;
      float nn = fmaxf(sqrtf(vx*vx + vy*vy + vz*vz), 1e-12f);
      nvx[t] = vx/nn; nvy[t] = vy/nn; nvz[t] = vz/nn;
    }
    float s = 0.f;
    #pragma unroll
    for (int k = 0; k < 6; ++k) {
      #pragma unroll
      for (int l = 0; l < 6; ++l) {
        float g = nvx[k]*nvx[l] + nvy[k]*nvy[l] + nvz[k]*nvz[l];
        g = fminf(fmaxf(g, -1.f + 1e-7f), 1.f - 1e-7f);
        s += expf(-acosf(g) * GRAD_INV_SIGMA);
      }
    }
    grad = s * (1.f/36.f);
  }
  grad = blockReduceSum(grad);
  if (threadIdx.x == 0) ws[PB + blockIdx.x] = grad;
}

__global__ void k_final(const float* __restrict__ ws, float* __restrict__ out,
                        int N, int M, int nbPrep, int nbA, int nbB)
{
  if (threadIdx.x != 0 || blockIdx.x != 0) return;
  float opS = 0.f, ent = 0.f, acS = 0.f, invS = 0.f, rat = 0.f;
  for (int b = 0; b < nbPrep; ++b) {
    opS  += ws[PP + 1*64 + b];
    ent  += ws[PP + 2*64 + b];
    acS  += ws[PP + 3*64 + b];
    invS += ws[PP + 4*64 + b];
    rat  += ws[PP + 5*64 + b];
  }
  float hub = 0.f, nrm = 0.f, con = 0.f, grd = 0.f;
  for (int b = 0; b < nbA; ++b) {
    hub += ws[PA + b]; nrm += ws[PA + 64 + b]; con += ws[PA + 128 + b];
  }
  for (int b = 0; b < nbB; ++b) grd += ws[PB + b];
  float Nf = (float)N, Mf = (float)M;
  float surface = (hub / Nf) * 1.0f;
  float normal  = (nrm / Nf) * 0.5f;
  float grad    = (grd / Mf) * 0.1f;
  float contact = (con / Nf) * 2.0f;
  float opacity = (opS / Mf + 0.1f * (ent / Mf)) * 0.01f;
  float scale   = (acS / (3.f*Mf) + invS / (3.f*Mf) + 0.1f * (rat / Mf)) * 0.001f;
  out[0] = surface + normal + grad + contact + opacity + scale;
}

extern "C" void kernel_launch(void* const* d_in, const int* in_sizes, int n_in,
                              void* d_out, int out_size, void* d_ws, size_t ws_size,
                              hipStream_t stream) {
  const float* tp  = (const float*)d_in[0];
  const float* tn  = (const float*)d_in[1];
  const float* tf  = (const float*)d_in[2];
  const float* pos = (const float*)d_in[3];
  const float* sc  = (const float*)d_in[4];
  const float* rot = (const float*)d_in[5];
  const float* op  = (const float*)d_in[6];
  float* ws = (float*)d_ws;
  float* out = (float*)d_out;

  const int N = in_sizes[0] / 3;   // 8192
  const int M = in_sizes[3] / 3;   // 8192
  const int nbPrep = ((N > M ? N : M) + 255) / 256;
  const int nbA = N / 128;
  const int nbB = M / 128;

  k_init<<<4, 256, 0, stream>>>(ws);
  k_prep<<<nbPrep, 256, 0, stream>>>(tf, pos, sc, op, ws, N, M);
  k_fmmean<<<1, 32, 0, stream>>>(ws, N, nbPrep);
  k_surface<<<nbA, 256, 0, stream>>>(tp, tn, tf, pos, sc, rot, ws, N, M);
  k_knn<<<nbB, 256, 0, stream>>>(pos, ws, M);
  k_final<<<1, 32, 0, stream>>>(ws, out, N, M, nbPrep, nbA, nbB);
}